// NMSLayer_27075473834342
// MI455X (gfx1250) — compile-verified
//
#include <hip/hip_runtime.h>
#include <hip/hip_bf16.h>
#include <cstdint>

// ---------------- problem constants ----------------
#define NCLS      80
#define NA1       8112        // 52*52*3
#define NA2       2028        // 26*26*3
#define NA3       507         // 13*13*3
#define NANCH     10647
#define NPAD      10656       // padded for alignment
#define TOPK      256
#define CAND_CAP  1024
#define MAXPC     100
#define MAXTOT    100
#define IOU_THRF  0.5f
#define SCORE_THRF 0.3f
#define NEGV      -1000000000.0f
#define NIMG      8
#define NFLAT     (NCLS * TOPK)   // 20480

// ---------------- CDNA5 async global->LDS DMA helpers ----------------
// GLOBAL_LOAD_ASYNC_TO_LDS_* (gfx1250): per-lane global address in a VGPR
// pair, per-lane LDS byte offset in a VGPR, tracked by ASYNCcnt.
__device__ __forceinline__ unsigned lds_off(const void* p) {
  // generic (flat) LDS address: low 32 bits are the wave-relative LDS offset
  return (unsigned)(uintptr_t)p;
}
__device__ __forceinline__ void async_load_b32_to_lds(unsigned dst, const void* src) {
  asm volatile("global_load_async_to_lds_b32 %0, %1, off"
               :: "v"(dst), "v"((unsigned long long)(uintptr_t)src)
               : "memory");
}
__device__ __forceinline__ void async_load_b128_to_lds(unsigned dst, const void* src) {
  asm volatile("global_load_async_to_lds_b128 %0, %1, off"
               :: "v"(dst), "v"((unsigned long long)(uintptr_t)src)
               : "memory");
}
__device__ __forceinline__ void wait_async0() {
  asm volatile("s_wait_asynccnt 0" ::: "memory");
}

// ---------------- kernel 1: per-(image,class) top-256 + NMS ----------------
// grid = (80 classes, 8 images), 256 threads (= 8 wave32).
__global__ void __launch_bounds__(256)
nms_per_class_kernel(const float* __restrict__ b1, const float* __restrict__ c1, const float* __restrict__ p1,
                     const float* __restrict__ b2, const float* __restrict__ c2, const float* __restrict__ p2,
                     const float* __restrict__ b3, const float* __restrict__ c3, const float* __restrict__ p3,
                     float* __restrict__ cls_scores, float* __restrict__ cls_boxes)
{
  const int cls = blockIdx.x;
  const int img = blockIdx.y;
  const int tid = threadIdx.x;

  extern __shared__ char smem[];
  float*              s_score = (float*)smem;                              // NPAD
  float*              s_conf  = s_score + NPAD;                            // NPAD
  unsigned*           s_hist  = (unsigned*)(s_conf + NPAD);                // 2048
  unsigned long long* s_cand  = (unsigned long long*)(s_hist + 2048);      // CAND_CAP
  unsigned long long* s_red   = s_cand + CAND_CAP;                         // 256
  unsigned long long* s_sel   = s_red + 256;                               // 256
  float*              s_y1    = (float*)(s_sel + 256);                     // 256 x6
  float*              s_x1    = s_y1 + TOPK;
  float*              s_y2    = s_x1 + TOPK;
  float*              s_x2    = s_y2 + TOPK;
  float*              s_ar    = s_x2 + TOPK;
  float*              s_sc    = s_ar + TOPK;
  int*                s_keep  = (int*)(s_sc + TOPK);                       // 256
  unsigned*           s_misc  = (unsigned*)(s_keep + TOPK);                // [0]=T [1]=cnt

  // ---- Phase A: async DMA the conf vector and the per-class prob column
  // (stride-80 gather) straight into LDS; overlap with histogram clear.
  for (int n = tid; n < NANCH; n += 256) {
    const float* cp; const float* pp;
    if (n < NA1)            { int m = img * NA1 + n;               cp = c1 + m; pp = p1 + (size_t)m * NCLS + cls; }
    else if (n < NA1 + NA2) { int m = img * NA2 + (n - NA1);       cp = c2 + m; pp = p2 + (size_t)m * NCLS + cls; }
    else                    { int m = img * NA3 + (n - NA1 - NA2); cp = c3 + m; pp = p3 + (size_t)m * NCLS + cls; }
    async_load_b32_to_lds(lds_off(&s_conf[n]),  cp);
    async_load_b32_to_lds(lds_off(&s_score[n]), pp);
  }
  for (int i = tid; i < 2048; i += 256) s_hist[i] = 0;   // disjoint LDS region
  wait_async0();
  __syncthreads();

  // ---- Phase B: score = conf*p; histogram on top-11 bits of the float key
  // (scores are positive => bit pattern order == numeric order).
  for (int n = tid; n < NANCH; n += 256) {
    float sc = s_conf[n] * s_score[n];
    s_score[n] = sc;
    if (sc > SCORE_THRF) {
      unsigned key = __float_as_uint(sc);
      atomicAdd(&s_hist[key >> 20], 1u);
    }
  }
  __syncthreads();

  // ---- Phase C: largest bucket T with suffix-count >= TOPK
  if (tid == 0) {
    unsigned acc = 0, T = 1;
    for (int h = 2047; h >= 1; --h) {
      acc += s_hist[h];
      if (acc >= TOPK) { T = (unsigned)h; break; }
    }
    s_misc[0] = T;
    s_misc[1] = 0;
  }
  __syncthreads();
  const unsigned T = s_misc[0];

  // ---- Phase D: compact survivors as (key<<32)|(N-n); larger composite ==
  // higher score, tie -> lower anchor index (matches jax top_k stability).
  for (int n = tid; n < NANCH; n += 256) {
    float sc = s_score[n];
    if (sc > SCORE_THRF) {
      unsigned key = __float_as_uint(sc);
      if ((key >> 20) >= T) {
        unsigned pos = atomicAdd(&s_misc[1], 1u);
        if (pos < CAND_CAP)
          s_cand[pos] = ((unsigned long long)key << 32) |
                        (unsigned long long)(unsigned)(NANCH - n);
      }
    }
  }
  __syncthreads();
  const unsigned M = (s_misc[1] < CAND_CAP) ? s_misc[1] : CAND_CAP;
  for (int i = tid; i < CAND_CAP; i += 256)
    if ((unsigned)i >= M) s_cand[i] = 0ull;
  __syncthreads();

  // ---- Phase E: exact ordered top-256 extraction over ~M (~300) candidates
  for (int i = 0; i < TOPK; ++i) {
    unsigned long long v  = s_cand[tid];
    unsigned long long v1 = s_cand[tid + 256]; if (v1 > v) v = v1;
    unsigned long long v2 = s_cand[tid + 512]; if (v2 > v) v = v2;
    unsigned long long v3 = s_cand[tid + 768]; if (v3 > v) v = v3;
    s_red[tid] = v;
    __syncthreads();
    for (int s = 128; s > 0; s >>= 1) {
      if (tid < s) { unsigned long long o = s_red[tid + s]; if (o > s_red[tid]) s_red[tid] = o; }
      __syncthreads();
    }
    unsigned long long w = s_red[0];
    if (tid == 0) s_sel[i] = w;
    if (w) {  // composites are unique; remove the winner
      if (s_cand[tid]       == w) s_cand[tid]       = 0ull;
      if (s_cand[tid + 256] == w) s_cand[tid + 256] = 0ull;
      if (s_cand[tid + 512] == w) s_cand[tid + 512] = 0ull;
      if (s_cand[tid + 768] == w) s_cand[tid + 768] = 0ull;
    }
    __syncthreads();
  }

  // ---- Phase F: gather the 256 selected boxes (one per thread)
  {
    unsigned long long w = s_sel[tid];
    float sc = NEGV; int n = -1;
    if (w) { sc = __uint_as_float((unsigned)(w >> 32)); n = NANCH - (int)(w & 0xffffffffull); }
    float4 bb = make_float4(0.f, 0.f, 0.f, 0.f);
    if (n >= 0) {
      const float* bp;
      if (n < NA1)            bp = b1 + (size_t)(img * NA1 + n) * 4;
      else if (n < NA1 + NA2) bp = b2 + (size_t)(img * NA2 + (n - NA1)) * 4;
      else                    bp = b3 + (size_t)(img * NA3 + (n - NA1 - NA2)) * 4;
      bb = *(const float4*)bp;
    }
    s_y1[tid] = bb.x; s_x1[tid] = bb.y; s_y2[tid] = bb.z; s_x2[tid] = bb.w;
    s_ar[tid] = (bb.z - bb.x) * (bb.w - bb.y);
    s_sc[tid] = sc;
    s_keep[tid] = 1;
  }
  __syncthreads();

  // ---- Phase G: sequential NMS (reference scan semantics)
  for (int i = 0; i < TOPK; ++i) {
    bool cur = (s_keep[i] != 0) && (s_sc[i] > SCORE_THRF);
    if (cur && tid > i && s_keep[tid]) {
      float iy = fminf(s_y2[i], s_y2[tid]) - fmaxf(s_y1[i], s_y1[tid]);
      float ix = fminf(s_x2[i], s_x2[tid]) - fmaxf(s_x1[i], s_x1[tid]);
      iy = fmaxf(iy, 0.f); ix = fmaxf(ix, 0.f);
      float inter = iy * ix;
      float uni = s_ar[i] + s_ar[tid] - inter;
      float iou = (uni > 0.f) ? (inter / uni) : 0.f;
      if (iou > IOU_THRF) s_keep[tid] = 0;
    }
    __syncthreads();
  }

  // ---- Phase H: cap at MAX_PER_CLASS, emit per-class slots
  if (tid == 0) {
    int cnt = 0;
    for (int i = 0; i < TOPK; ++i) {
      int k = (s_keep[i] != 0) && (s_sc[i] > SCORE_THRF);
      if (k) { ++cnt; if (cnt > MAXPC) k = 0; }
      s_keep[i] = k;
    }
  }
  __syncthreads();
  const size_t o = ((size_t)img * NCLS + cls) * TOPK + tid;
  cls_scores[o] = s_keep[tid] ? s_sc[tid] : NEGV;
  ((float4*)cls_boxes)[o] = make_float4(s_y1[tid], s_x1[tid], s_y2[tid], s_x2[tid]);
}

// ---------------- kernel 2: per-image top-100 combine ----------------
__global__ void __launch_bounds__(256)
nms_combine_kernel(const float* __restrict__ cls_scores,
                   const float* __restrict__ cls_boxes,
                   float* __restrict__ out)
{
  const int img = blockIdx.x;
  const int tid = threadIdx.x;

  extern __shared__ char smem[];
  float* s_s  = (float*)smem;            // NFLAT
  float* s_rv = s_s + NFLAT;             // 256
  int*   s_ri = (int*)(s_rv + 256);      // 256

  // bulk async DMA of the per-image flat score array into LDS (B128)
  const float* src = cls_scores + (size_t)img * NFLAT;
  for (int i = tid * 4; i < NFLAT; i += 256 * 4)
    async_load_b128_to_lds(lds_off(&s_s[i]), src + i);
  wait_async0();
  __syncthreads();

  float* out_boxes   = out;
  float* out_scores  = out + (size_t)NIMG * MAXTOT * 4;
  float* out_classes = out_scores + (size_t)NIMG * MAXTOT;
  float* out_nvalid  = out_classes + (size_t)NIMG * MAXTOT;

  int nv = 0;  // thread 0 accumulates across iterations
  for (int k = 0; k < MAXTOT; ++k) {
    float bv = -3.402823466e38f; int bi = NFLAT;
    for (int i = tid; i < NFLAT; i += 256) {
      float v = s_s[i];
      if (v > bv) { bv = v; bi = i; }     // strict > keeps lowest index
    }
    s_rv[tid] = bv; s_ri[tid] = bi;
    __syncthreads();
    for (int s = 128; s > 0; s >>= 1) {
      if (tid < s) {
        float ov = s_rv[tid + s]; int oi = s_ri[tid + s];
        if (ov > s_rv[tid] || (ov == s_rv[tid] && oi < s_ri[tid])) { s_rv[tid] = ov; s_ri[tid] = oi; }
      }
      __syncthreads();
    }
    if (tid == 0) {
      float sc = s_rv[0]; int fi = s_ri[0];
      bool valid = (fi < NFLAT) && (sc > SCORE_THRF);
      float4 bb = make_float4(0.f, 0.f, 0.f, 0.f);
      if (valid) {
        bb = ((const float4*)cls_boxes)[(size_t)img * NFLAT + fi];
        bb.x = fminf(fmaxf(bb.x, 0.f), 1.f);
        bb.y = fminf(fmaxf(bb.y, 0.f), 1.f);
        bb.z = fminf(fmaxf(bb.z, 0.f), 1.f);
        bb.w = fminf(fmaxf(bb.w, 0.f), 1.f);
      }
      ((float4*)out_boxes)[(size_t)img * MAXTOT + k] = bb;
      out_scores[img * MAXTOT + k]  = valid ? sc : 0.f;
      out_classes[img * MAXTOT + k] = valid ? (float)(fi / TOPK) : 0.f;
      if (valid) ++nv;
      if (fi < NFLAT) s_s[fi] = -3.402823466e38f;  // remove winner
    }
    __syncthreads();
  }
  if (tid == 0) out_nvalid[img] = (float)nv;
}

// ---------------- launcher ----------------
extern "C" void kernel_launch(void* const* d_in, const int* in_sizes, int n_in,
                              void* d_out, int out_size, void* d_ws, size_t ws_size,
                              hipStream_t stream) {
  (void)in_sizes; (void)n_in; (void)out_size; (void)ws_size;
  const float* b1 = (const float*)d_in[0];
  const float* c1 = (const float*)d_in[1];
  const float* p1 = (const float*)d_in[2];
  const float* b2 = (const float*)d_in[3];
  const float* c2 = (const float*)d_in[4];
  const float* p2 = (const float*)d_in[5];
  const float* b3 = (const float*)d_in[6];
  const float* c3 = (const float*)d_in[7];
  const float* p3 = (const float*)d_in[8];

  float* cls_scores = (float*)d_ws;                              // 8*80*256 f32
  float* cls_boxes  = cls_scores + (size_t)NIMG * NCLS * TOPK;   // 8*80*256*4 f32 (16B aligned)

  const size_t sh2 =
      (size_t)NPAD * 4 * 2           // s_score + s_conf
    + 2048 * 4                       // histogram
    + (size_t)CAND_CAP * 8           // candidates
    + 256 * 8 + 256 * 8              // reduce + selected
    + 6 * 256 * 4                    // y1,x1,y2,x2,area,score
    + 256 * 4                        // keep
    + 2 * 4;                         // misc
  dim3 g2(NCLS, NIMG);
  nms_per_class_kernel<<<g2, 256, sh2, stream>>>(b1, c1, p1, b2, c2, p2, b3, c3, p3,
                                                 cls_scores, cls_boxes);

  const size_t sh3 = (size_t)NFLAT * 4 + 256 * 4 + 256 * 4;
  nms_combine_kernel<<<NIMG, 256, sh3, stream>>>(cls_scores, cls_boxes, (float*)d_out);
}